// GCN_12910671691940
// MI455X (gfx1250) — compile-verified
//
#include <hip/hip_runtime.h>
#include <hip/hip_bf16.h>
#include <math.h>

// ---------------- problem constants (match reference) ----------------
#define NN    50000
#define EE    800000
#define DD    128
#define HH    6
#define CC    64
#define HC    384      // H * C
#define GED   4
#define EAD   2
#define NCLS  10
#define NG    256
#define KHOP  5
#define ALPHA 0.1f

typedef float v2f __attribute__((ext_vector_type(2)));
typedef float v8f __attribute__((ext_vector_type(8)));

// ---------------- helpers ----------------
__device__ __forceinline__ float sigm(float x) { return 1.0f / (1.0f + __expf(-x)); }
__device__ __forceinline__ float lrelu(float x) { return x > 0.0f ? x : 0.2f * x; }

// order-preserving float<->uint encoding for atomic max on floats
__device__ __forceinline__ unsigned encf(float f) {
    unsigned u = __float_as_uint(f);
    return (u & 0x80000000u) ? ~u : (u | 0x80000000u);
}
__device__ __forceinline__ float decf(unsigned u) {
    unsigned b = (u & 0x80000000u) ? (u & 0x7FFFFFFFu) : ~u;
    return __uint_as_float(b);
}
#define NEG_INF_ENC 0x007FFFFFu   // encf(-inf)

// ---------------- edge MLPs: ew, ea1, ea2 ----------------
__global__ void k_edge_mlps(const float* __restrict__ ea,
                            const float* __restrict__ eww1, const float* __restrict__ ewb1,
                            const float* __restrict__ eww2, const float* __restrict__ ewb2,
                            const float* __restrict__ a1w1, const float* __restrict__ a1b1,
                            const float* __restrict__ a1w2, const float* __restrict__ a1b2,
                            const float* __restrict__ a2w1, const float* __restrict__ a2b1,
                            const float* __restrict__ a2w2, const float* __restrict__ a2b2,
                            float* __restrict__ ew, float* __restrict__ ea1, float* __restrict__ ea2) {
    int e = blockIdx.x * blockDim.x + threadIdx.x;
    if (e >= EE) return;
    float x0 = ea[2 * e], x1 = ea[2 * e + 1];
    // ew = sigmoid( sigmoid(x@w1+b1) @ w2 + b2 )
    float z = ewb2[0];
#pragma unroll
    for (int j = 0; j < 4; ++j) {
        float h = sigm(x0 * eww1[j] + x1 * eww1[4 + j] + ewb1[j]);
        z += h * eww2[j];
    }
    ew[e] = sigm(z);
    // ea1 = relu(x@w1+b1)@w2+b2   (2->8->4)
    float h1[8], h2[8];
#pragma unroll
    for (int j = 0; j < 8; ++j) {
        h1[j] = fmaxf(x0 * a1w1[j] + x1 * a1w1[8 + j] + a1b1[j], 0.0f);
        h2[j] = fmaxf(x0 * a2w1[j] + x1 * a2w1[8 + j] + a2b1[j], 0.0f);
    }
#pragma unroll
    for (int g = 0; g < GED; ++g) {
        float s1 = a1b2[g], s2 = a2b2[g];
#pragma unroll
        for (int j = 0; j < 8; ++j) {
            s1 += h1[j] * a1w2[j * GED + g];
            s2 += h2[j] * a2w2[j * GED + g];
        }
        ea1[e * GED + g] = s1;
        ea2[e * GED + g] = s2;
    }
}

// ---------------- APPNP norm ----------------
__global__ void k_deg(const int* __restrict__ dst, const float* __restrict__ ew,
                      float* __restrict__ deg) {
    int e = blockIdx.x * blockDim.x + threadIdx.x;
    if (e >= EE) return;
    atomicAdd(&deg[dst[e]], ew[e]);
}
__global__ void k_dinv(float* __restrict__ deg) {
    int i = blockIdx.x * blockDim.x + threadIdx.x;
    if (i >= NN) return;
    deg[i] = rsqrtf(deg[i] + 1.0f);   // +1 for self loop, always > 0
}
__global__ void k_norm(const int* __restrict__ src, const int* __restrict__ dst,
                       const float* __restrict__ ew, const float* __restrict__ dinv,
                       float* __restrict__ norm) {
    int e = blockIdx.x * blockDim.x + threadIdx.x;
    if (e >= EE) return;
    norm[e] = dinv[src[e]] * ew[e] * dinv[dst[e]];
}

// ---------------- APPNP propagation ----------------
__global__ void k_self_init(const float* __restrict__ h, const float* __restrict__ dinv,
                            float* __restrict__ tgt) {
    int idx = blockIdx.x * blockDim.x + threadIdx.x;
    if (idx >= NN * DD) return;
    float s = dinv[idx / DD];
    tgt[idx] = s * s * h[idx];
}
__global__ void k_edge_scatter(const int* __restrict__ src, const int* __restrict__ dst,
                               const float* __restrict__ norm, const float* __restrict__ h,
                               float* __restrict__ tgt) {
    const int vecs = DD / 4;
    long idx = (long)blockIdx.x * blockDim.x + threadIdx.x;
    if (idx >= (long)EE * vecs) return;
    int e = (int)(idx / vecs);
    int q = (int)(idx % vecs) * 4;
    int s = src[e], t = dst[e];
    // gfx1250: prefetch source row of edge e+8 into cache
    int en = e + 8 < EE ? e + 8 : e;
    __builtin_prefetch(h + (size_t)src[en] * DD + q, 0, 1);
    float w = norm[e];
    const float4 v = *reinterpret_cast<const float4*>(h + (size_t)s * DD + q);
    float* o = tgt + (size_t)t * DD + q;
    atomicAdd(o + 0, w * v.x);
    atomicAdd(o + 1, w * v.y);
    atomicAdd(o + 2, w * v.z);
    atomicAdd(o + 3, w * v.w);
}
__global__ void k_combine(float* __restrict__ tgt, const float* __restrict__ x0) {
    int idx = blockIdx.x * blockDim.x + threadIdx.x;
    if (idx >= NN * DD) return;
    tgt[idx] = (1.0f - ALPHA) * tgt[idx] + ALPHA * x0[idx];
}
__global__ void k_relu(float* __restrict__ x, int n) {
    int idx = blockIdx.x * blockDim.x + threadIdx.x;
    if (idx >= n) return;
    x[idx] = fmaxf(x[idx], 0.0f);
}

// ---------------- fp32 WMMA GEMM:  C[n,m] = A[n,k] @ W[k,m] ----------------
// one wave per 16x16 C-tile; V_WMMA_F32_16X16X4_F32, K step 4
__global__ void k_gemm_wmma_f32(const float* __restrict__ A, const float* __restrict__ W,
                                float* __restrict__ C, int n, int k, int m) {
    const int wave = threadIdx.x >> 5;
    const int lane = threadIdx.x & 31;
    const int tilesN = m >> 4;
    const int tile = blockIdx.x * (blockDim.x >> 5) + wave;
    if (tile >= (n >> 4) * tilesN) return;        // wave-uniform guard (EXEC stays all-1s)
    const int tm = (tile / tilesN) << 4;
    const int tn = (tile % tilesN) << 4;
    const int r  = lane & 15;
    const int kh = (lane >> 4) << 1;              // lanes 0-15: K=0,1  lanes 16-31: K=2,3
    v8f acc = {};
    for (int kb = 0; kb < k; kb += 4) {
        v2f a, b;
        a.x = A[(size_t)(tm + r) * k + kb + kh];
        a.y = A[(size_t)(tm + r) * k + kb + kh + 1];
        b.x = W[(size_t)(kb + kh) * m + tn + r];
        b.y = W[(size_t)(kb + kh + 1) * m + tn + r];
        acc = __builtin_amdgcn_wmma_f32_16x16x4_f32(false, a, false, b, (short)0, acc,
                                                    false, false);
    }
    const int rbase = tm + ((lane >> 4) << 3);    // VGPR i -> row i (lanes 0-15) / 8+i (16-31)
#pragma unroll
    for (int i = 0; i < 8; ++i)
        C[(size_t)(rbase + i) * m + tn + r] = acc[i];
}

// ---------------- GAT pieces ----------------
__global__ void k_attdot(const float* __restrict__ xs, const float* __restrict__ att,
                         float* __restrict__ out) {   // out[i*H+h] = <xs[i,h,:], att[h,:]>
    int idx = blockIdx.x * blockDim.x + threadIdx.x;
    if (idx >= NN * HH) return;
    int i = idx / HH, h = idx % HH;
    const float* xr = xs + (size_t)i * HC + h * CC;
    const float* ar = att + h * CC;
    float s = 0.0f;
#pragma unroll 8
    for (int j = 0; j < CC; ++j) s += xr[j] * ar[j];
    out[idx] = s;
}
__global__ void k_weatt(const float* __restrict__ wedge, const float* __restrict__ atte,
                        float* __restrict__ weatt) {  // weatt[g*H+h]
    int t = threadIdx.x;
    if (t >= GED * HH) return;
    int g = t / HH, h = t % HH;
    float s = 0.0f;
#pragma unroll 8
    for (int j = 0; j < CC; ++j) s += wedge[(size_t)g * HC + h * CC + j] * atte[h * CC + j];
    weatt[t] = s;
}
__global__ void k_loopea_scatter(const int* __restrict__ dst, const float* __restrict__ ea,
                                 float* __restrict__ lsum, float* __restrict__ cnt) {
    int e = blockIdx.x * blockDim.x + threadIdx.x;
    if (e >= EE) return;
    int t = dst[e];
    atomicAdd(&cnt[t], 1.0f);
#pragma unroll
    for (int g = 0; g < GED; ++g) atomicAdd(&lsum[t * GED + g], ea[e * GED + g]);
}
__global__ void k_loopea_final(const float* __restrict__ lsum, const float* __restrict__ cnt,
                               float* __restrict__ lea) {
    int idx = blockIdx.x * blockDim.x + threadIdx.x;
    if (idx >= NN * GED) return;
    lea[idx] = lsum[idx] / fmaxf(cnt[idx / GED], 1.0f);
}
__global__ void k_fill_u32(unsigned* __restrict__ p, unsigned v, int n) {
    int idx = blockIdx.x * blockDim.x + threadIdx.x;
    if (idx < n) p[idx] = v;
}
__global__ void k_alpha_max(const int* __restrict__ src, const int* __restrict__ dst,
                            const float* __restrict__ ea, const float* __restrict__ lea,
                            const float* __restrict__ als, const float* __restrict__ ald,
                            const float* __restrict__ weatt,
                            float* __restrict__ alpha, unsigned* __restrict__ segm) {
    long idx = (long)blockIdx.x * blockDim.x + threadIdx.x;
    if (idx >= (long)(EE + NN) * HH) return;
    int e = (int)(idx / HH), h = (int)(idx % HH);
    int s, t;
    const float* eav;
    if (e < EE) { s = src[e]; t = dst[e]; eav = ea + (size_t)e * GED; }
    else        { s = t = e - EE;          eav = lea + (size_t)s * GED; }
    float ale = 0.0f;
#pragma unroll
    for (int g = 0; g < GED; ++g) ale += eav[g] * weatt[g * HH + h];
    float a = lrelu(als[s * HH + h] + ald[t * HH + h] + ale);
    alpha[idx] = a;
    atomicMax(&segm[t * HH + h], encf(a));
}
__global__ void k_exp_sum(const int* __restrict__ dst, const unsigned* __restrict__ segm,
                          float* __restrict__ alpha, float* __restrict__ segs) {
    long idx = (long)blockIdx.x * blockDim.x + threadIdx.x;
    if (idx >= (long)(EE + NN) * HH) return;
    int e = (int)(idx / HH), h = (int)(idx % HH);
    int t = (e < EE) ? dst[e] : (e - EE);
    float m  = decf(segm[t * HH + h]);
    float ex = __expf(alpha[idx] - m);
    alpha[idx] = ex;
    atomicAdd(&segs[t * HH + h], ex);
}
__global__ void k_gat_scatter(const int* __restrict__ src, const int* __restrict__ dst,
                              const float* __restrict__ alpha, const float* __restrict__ segs,
                              const float* __restrict__ xs, float* __restrict__ outacc) {
    const int per = HH * (CC / 4);                    // 96 quads per edge
    long idx = (long)blockIdx.x * blockDim.x + threadIdx.x;
    if (idx >= (long)(EE + NN) * per) return;
    int e   = (int)(idx / per);
    int rem = (int)(idx % per);
    int h   = rem / (CC / 4);
    int q   = (rem % (CC / 4)) * 4;
    int s, t;
    if (e < EE) { s = src[e]; t = dst[e]; }
    else        { s = t = e - EE; }
    float w = alpha[(size_t)e * HH + h] / (segs[t * HH + h] + 1e-16f);
    const float4 v = *reinterpret_cast<const float4*>(xs + (size_t)s * HC + h * CC + q);
    float* o = outacc + (size_t)t * HC + h * CC + q;
    atomicAdd(o + 0, w * v.x);
    atomicAdd(o + 1, w * v.y);
    atomicAdd(o + 2, w * v.z);
    atomicAdd(o + 3, w * v.w);
}
__global__ void k_gat_final(const float* __restrict__ outacc, const float* __restrict__ bias,
                            float* __restrict__ out) {  // head mean + bias + relu
    int idx = blockIdx.x * blockDim.x + threadIdx.x;
    if (idx >= NN * CC) return;
    int i = idx / CC, c = idx % CC;
    float s = 0.0f;
#pragma unroll
    for (int h = 0; h < HH; ++h) s += outacc[(size_t)i * HC + h * CC + c];
    out[idx] = fmaxf(s * (1.0f / HH) + bias[c], 0.0f);
}

// ---------------- pooling + classifier ----------------
__global__ void k_pool(const int* __restrict__ batch, const float* __restrict__ h,
                       float* __restrict__ pool, float* __restrict__ pcnt) {
    int idx = blockIdx.x * blockDim.x + threadIdx.x;
    if (idx >= NN * CC) return;
    int i = idx / CC, c = idx % CC;
    int b = batch[i];
    atomicAdd(&pool[b * CC + c], h[idx]);
    if (c == 0) atomicAdd(&pcnt[b], 1.0f);
}
__global__ void k_final(const float* __restrict__ pool, const float* __restrict__ pcnt,
                        const float* __restrict__ lw, const float* __restrict__ lb,
                        float* __restrict__ out) {
    int g = blockIdx.x * blockDim.x + threadIdx.x;
    if (g >= NG) return;
    float inv = 1.0f / fmaxf(pcnt[g], 1.0f);
    float l[NCLS];
#pragma unroll
    for (int j = 0; j < NCLS; ++j) l[j] = lb[j];
    for (int c = 0; c < CC; ++c) {
        float gv = fmaxf(pool[g * CC + c] * inv, 0.0f);
#pragma unroll
        for (int j = 0; j < NCLS; ++j) l[j] += gv * lw[c * NCLS + j];
    }
    float m = l[0];
#pragma unroll
    for (int j = 1; j < NCLS; ++j) m = fmaxf(m, l[j]);
    float s = 0.0f;
#pragma unroll
    for (int j = 0; j < NCLS; ++j) s += __expf(l[j] - m);
    float lse = logf(s);
#pragma unroll
    for (int j = 0; j < NCLS; ++j) out[g * NCLS + j] = l[j] - m - lse;
}

// ---------------- host orchestration ----------------
static inline unsigned nblk(long total, int bs) { return (unsigned)((total + bs - 1) / bs); }

extern "C" void kernel_launch(void* const* d_in, const int* in_sizes, int n_in,
                              void* d_out, int out_size, void* d_ws, size_t ws_size,
                              hipStream_t stream) {
    // --- inputs (setup_inputs dict order, params flattened in insertion order) ---
    const float* x      = (const float*)d_in[0];           // [N,128]
    const int*   eidx   = (const int*)d_in[1];             // [2,E]
    const float* eattr  = (const float*)d_in[2];           // [E,2]
    const int*   batch  = (const int*)d_in[3];             // [N]
    const float* ew_w1  = (const float*)d_in[4];
    const float* ew_b1  = (const float*)d_in[5];
    const float* ew_w2  = (const float*)d_in[6];
    const float* ew_b2  = (const float*)d_in[7];
    const float* ea1_w1 = (const float*)d_in[8];
    const float* ea1_b1 = (const float*)d_in[9];
    const float* ea1_w2 = (const float*)d_in[10];
    const float* ea1_b2 = (const float*)d_in[11];
    const float* ea2_w1 = (const float*)d_in[12];
    const float* ea2_b1 = (const float*)d_in[13];
    const float* ea2_w2 = (const float*)d_in[14];
    const float* ea2_b2 = (const float*)d_in[15];
    const float* g1_wsrc = (const float*)d_in[16];
    const float* g1_wdst = (const float*)d_in[17];
    const float* g1_wedge = (const float*)d_in[18];
    const float* g1_asrc = (const float*)d_in[19];
    const float* g1_adst = (const float*)d_in[20];
    const float* g1_aedge = (const float*)d_in[21];
    const float* g1_bias = (const float*)d_in[22];
    const float* g2_wsrc = (const float*)d_in[23];
    const float* g2_wdst = (const float*)d_in[24];
    const float* g2_wedge = (const float*)d_in[25];
    const float* g2_asrc = (const float*)d_in[26];
    const float* g2_adst = (const float*)d_in[27];
    const float* g2_aedge = (const float*)d_in[28];
    const float* g2_bias = (const float*)d_in[29];
    const float* lin_w  = (const float*)d_in[30];
    const float* lin_b  = (const float*)d_in[31];

    const int* src = eidx;
    const int* dst = eidx + EE;

    // --- workspace arena (floats), 256B-aligned slots ---
    float* W = (float*)d_ws;
    size_t off = 0;
    auto alloc = [&](size_t cnt) -> float* {
        float* p = W + off;
        off += (cnt + 63) & ~(size_t)63;
        return p;
    };
    float* ew      = alloc(EE);
    float* ea1     = alloc((size_t)EE * GED);
    float* ea2     = alloc((size_t)EE * GED);
    float* dinv    = alloc(NN);                 // deg -> dinv in place
    float* normw   = alloc(EE);
    float* hA      = alloc((size_t)NN * DD);
    float* hB      = alloc((size_t)NN * DD);    // also reused as GAT1 output (N*64)
    float* xs      = alloc((size_t)NN * HC);
    float* xd      = alloc((size_t)NN * HC);
    float* outacc  = alloc((size_t)NN * HC);
    float* als     = alloc((size_t)NN * HH);
    float* ald     = alloc((size_t)NN * HH);
    float* lsum    = alloc((size_t)NN * GED);
    float* cnt     = alloc(NN);
    float* lea     = alloc((size_t)NN * GED);
    float* alpha   = alloc((size_t)(EE + NN) * HH);
    unsigned* segm = (unsigned*)alloc((size_t)NN * HH);
    float* segs    = alloc((size_t)NN * HH);
    float* weatt   = alloc(GED * HH);
    float* pool    = alloc((size_t)NG * CC);
    float* pcnt    = alloc(NG);
    (void)ws_size; (void)n_in; (void)in_sizes; (void)out_size;

    const int BS = 256;

    // === Stage A: edge MLPs ===
    k_edge_mlps<<<nblk(EE, BS), BS, 0, stream>>>(eattr,
        ew_w1, ew_b1, ew_w2, ew_b2,
        ea1_w1, ea1_b1, ea1_w2, ea1_b2,
        ea2_w1, ea2_b1, ea2_w2, ea2_b2,
        ew, ea1, ea2);

    // === Stage B: gcn_norm ===
    hipMemsetAsync(dinv, 0, NN * sizeof(float), stream);
    k_deg<<<nblk(EE, BS), BS, 0, stream>>>(dst, ew, dinv);
    k_dinv<<<nblk(NN, BS), BS, 0, stream>>>(dinv);
    k_norm<<<nblk(EE, BS), BS, 0, stream>>>(src, dst, ew, dinv, normw);

    // === Stage C: APPNP, K hops (h starts at x; ping-pong hA/hB) ===
    const float* cur = x;
    float* bufs[2] = {hA, hB};
    for (int hop = 0; hop < KHOP; ++hop) {
        float* tgt = bufs[hop & 1];               // hop0->hA, hop1->hB, ... hop4->hA
        k_self_init<<<nblk((long)NN * DD, BS), BS, 0, stream>>>(cur, dinv, tgt);
        k_edge_scatter<<<nblk((long)EE * (DD / 4), BS), BS, 0, stream>>>(src, dst, normw, cur, tgt);
        k_combine<<<nblk((long)NN * DD, BS), BS, 0, stream>>>(tgt, x);
        cur = tgt;
    }
    k_relu<<<nblk((long)NN * DD, BS), BS, 0, stream>>>(hA, NN * DD);   // final hop landed in hA

    // === GAT layers (shared scratch, re-zeroed per layer) ===
    const int gemmTiles = (NN / 16) * (HC / 16);       // 3125*24 = 75000, /8 waves = 9375 blocks
    struct GatCfg {
        const float* hin; int fi;
        const float* wsrc; const float* wdst; const float* wedge;
        const float* asrc; const float* adst; const float* aedge; const float* bias;
        const float* eaL; float* hout;
    };
    GatCfg layers[2] = {
        { hA, DD, g1_wsrc, g1_wdst, g1_wedge, g1_asrc, g1_adst, g1_aedge, g1_bias, ea1, hB },
        { hB, CC, g2_wsrc, g2_wdst, g2_wedge, g2_asrc, g2_adst, g2_aedge, g2_bias, ea2, hA },
    };
    for (int L = 0; L < 2; ++L) {
        GatCfg& g = layers[L];
        // dense projections via fp32 WMMA
        k_gemm_wmma_f32<<<gemmTiles / 8, 256, 0, stream>>>(g.hin, g.wsrc, xs, NN, g.fi, HC);
        k_gemm_wmma_f32<<<gemmTiles / 8, 256, 0, stream>>>(g.hin, g.wdst, xd, NN, g.fi, HC);
        // attention logit pieces
        k_attdot<<<nblk((long)NN * HH, BS), BS, 0, stream>>>(xs, g.asrc, als);
        k_attdot<<<nblk((long)NN * HH, BS), BS, 0, stream>>>(xd, g.adst, ald);
        k_weatt<<<1, 32, 0, stream>>>(g.wedge, g.aedge, weatt);
        // self-loop edge attr = scatter-mean by dst
        hipMemsetAsync(lsum, 0, (size_t)NN * GED * sizeof(float), stream);
        hipMemsetAsync(cnt,  0, NN * sizeof(float), stream);
        k_loopea_scatter<<<nblk(EE, BS), BS, 0, stream>>>(dst, g.eaL, lsum, cnt);
        k_loopea_final<<<nblk((long)NN * GED, BS), BS, 0, stream>>>(lsum, cnt, lea);
        // alpha + segment softmax
        k_fill_u32<<<nblk((long)NN * HH, BS), BS, 0, stream>>>(segm, NEG_INF_ENC, NN * HH);
        hipMemsetAsync(segs, 0, (size_t)NN * HH * sizeof(float), stream);
        k_alpha_max<<<nblk((long)(EE + NN) * HH, BS), BS, 0, stream>>>(
            src, dst, g.eaL, lea, als, ald, weatt, alpha, segm);
        k_exp_sum<<<nblk((long)(EE + NN) * HH, BS), BS, 0, stream>>>(dst, segm, alpha, segs);
        // weighted scatter of xs[src] into outacc[dst]
        hipMemsetAsync(outacc, 0, (size_t)NN * HC * sizeof(float), stream);
        k_gat_scatter<<<nblk((long)(EE + NN) * HH * (CC / 4), BS), BS, 0, stream>>>(
            src, dst, alpha, segs, xs, outacc);
        // head mean + bias + relu
        k_gat_final<<<nblk((long)NN * CC, BS), BS, 0, stream>>>(outacc, g.bias, g.hout);
    }

    // === pooling + classifier ===
    hipMemsetAsync(pool, 0, (size_t)NG * CC * sizeof(float), stream);
    hipMemsetAsync(pcnt, 0, NG * sizeof(float), stream);
    k_pool<<<nblk((long)NN * CC, BS), BS, 0, stream>>>(batch, hA, pool, pcnt);
    k_final<<<1, NG, 0, stream>>>(pool, pcnt, lin_w, lin_b, (float*)d_out);
}